// CNN_BiMACL_31860067401819
// MI455X (gfx1250) — compile-verified
//
#include <hip/hip_runtime.h>
#include <math.h>

// ---------------- static problem config (mirrors reference) ----------------
#define WAY   5
#define SHOT  16
#define SEQL  10
#define DIN   2048
#define DOUT  1152
#define NQ    320
#define TT    45            // C(10,2)
#define SSUP  720           // SHOT*TT (support tuples per class)
#define KDIM  4096          // TSS*DIN
#define MQ    (NQ*TT)       // 14400 query-tuple rows
#define MS    (WAY*SSUP)    // 3600 support-tuple rows
#define MTOT  (MQ+MS)       // 18000

// itertools.combinations(range(10), 2) in order
__constant__ int c_tup0[TT] = {
  0,0,0,0,0,0,0,0,0,
  1,1,1,1,1,1,1,1,
  2,2,2,2,2,2,2,
  3,3,3,3,3,3,
  4,4,4,4,4,
  5,5,5,5,
  6,6,6,
  7,7,
  8};
__constant__ int c_tup1[TT] = {
  1,2,3,4,5,6,7,8,9,
  2,3,4,5,6,7,8,9,
  3,4,5,6,7,8,9,
  4,5,6,7,8,9,
  5,6,7,8,9,
  6,7,8,9,
  7,8,9,
  8,9,
  9};

typedef __attribute__((ext_vector_type(2))) float v2f;
typedef __attribute__((ext_vector_type(8))) float v8f;

__device__ __forceinline__ v8f wmma_f32(v2f a, v2f b, v8f c) {
  // V_WMMA_F32_16X16X4_F32 : D = A(16x4,f32) * B(4x16,f32) + C(16x16,f32)
  return __builtin_amdgcn_wmma_f32_16x16x4_f32(
      /*neg_a=*/false, a, /*neg_b=*/false, b,
      /*c_mod=*/(short)0, c, /*reuse_a=*/false, /*reuse_b=*/false);
}

__device__ __forceinline__ v2f mk2(float x, float y) { v2f p; p.x = x; p.y = y; return p; }

// ============================================================================
// Kernel 1: E[m, n] = relu( X[m,:] . W[n,:] + b[n] ),  m in [0,18000)
//   X rows are gathered frame-pairs from queries (m < MQ) or support (m >= MQ).
//   Macrotile 128(M) x 64(N), 8 waves in a 4x2 grid, each wave 32x32 output
//   (4 f32 WMMA accumulators). LDS double-buffered: ONE barrier per K-stage.
//   LDS holds K-PAIRED v2f elements so each WMMA operand is a single aligned
//   ds_load_b64 (no VGPR glue): operand = buf[kk/2 + laneHalf][row/col].
// ============================================================================
__global__ __launch_bounds__(256)
void k_embed(const float* __restrict__ sup, const float* __restrict__ qry,
             const float* __restrict__ W,   const float* __restrict__ bias,
             float* __restrict__ E)
{
  __shared__ v2f As[2][8][130];   // [buf][kpair][m] ; element = {A[2kp], A[2kp+1]}
  __shared__ v2f Bs[2][8][66];    // [buf][kpair][n]
  const int tid   = threadIdx.x;
  const int mTile = blockIdx.x * 128;
  const int nTile = blockIdx.y * 64;

  // --- A loader: thread -> (row, 8 consecutive k = pairs kp0..kp0+3) ---
  const int aRow = tid >> 1;          // 0..127
  const int aKp  = (tid & 1) * 4;     // pair base: 0 or 4 (k = 0 or 8)
  int m  = mTile + aRow;
  int mc = (m < MTOT) ? m : 0;        // clamp OOB rows (store-guarded later)
  const float *base0, *base1;
  if (mc < MQ) {
    int q = mc / TT, t = mc % TT;
    base0 = qry + (size_t)(q*SEQL + c_tup0[t]) * DIN;
    base1 = qry + (size_t)(q*SEQL + c_tup1[t]) * DIN;
  } else {
    int rs = mc - MQ;
    int cc = rs / SSUP, rem = rs % SSUP;
    int sh = rem / TT,  t  = rem % TT;
    int si = cc*SHOT + sh;            // labels sorted -> identity grouping
    base0 = sup + (size_t)(si*SEQL + c_tup0[t]) * DIN;
    base1 = sup + (size_t)(si*SEQL + c_tup1[t]) * DIN;
  }

  // --- B loader: thread -> (col, pairs bKp..bKp+1) ; B[k][n] = W[n][k] ---
  const int bN  = tid & 63;
  const int bKp = (tid >> 6) * 2;     // 0,2,4,6  (k = 4*bKp/2)
  const float* wrow = W + (size_t)(nTile + bN) * KDIM;  // DOUT%64==0, no guard

  // --- wave tiling: 4 (M) x 2 (N) waves, each 32x32 ---
  const int wv   = tid >> 5;
  const int lane = tid & 31;
  const int lm   = lane & 15, lh = lane >> 4;
  const int mw   = (wv & 3) * 32;
  const int nw   = (wv >> 2) * 32;

  v8f acc00 = {}, acc01 = {}, acc10 = {}, acc11 = {};

  const int NSTAGE = KDIM / 16;       // 256 (each stage entirely in one frame)

  // prologue: stage 0 -> buf 0
  {
    const float* ab = base0;          // k0 = 0 < DIN
    float4 ar0 = *(const float4*)(ab + aKp*2);
    float4 ar1 = *(const float4*)(ab + aKp*2 + 4);
    float4 br  = *(const float4*)(wrow + bKp*2);
    As[0][aKp+0][aRow] = mk2(ar0.x, ar0.y);
    As[0][aKp+1][aRow] = mk2(ar0.z, ar0.w);
    As[0][aKp+2][aRow] = mk2(ar1.x, ar1.y);
    As[0][aKp+3][aRow] = mk2(ar1.z, ar1.w);
    Bs[0][bKp+0][bN]   = mk2(br.x, br.y);
    Bs[0][bKp+1][bN]   = mk2(br.z, br.w);
  }
  __syncthreads();

  for (int s = 0; s < NSTAGE; ++s) {
    const int cur = s & 1;
    float4 ar0, ar1, br;
    const bool more = (s + 1) < NSTAGE;
    if (more) {
      int k1 = (s + 1) * 16;          // 2048 % 16 == 0 -> stage in one frame
      const float* ab = (k1 < DIN) ? (base0 + k1) : (base1 + (k1 - DIN));
      ar0 = *(const float4*)(ab + aKp*2);
      ar1 = *(const float4*)(ab + aKp*2 + 4);
      br  = *(const float4*)(wrow + k1 + bKp*2);
    }
#pragma unroll
    for (int kk2 = 0; kk2 < 8; kk2 += 2) {     // kk2 = kpair base of 4-wide K step
      v2f a0 = As[cur][kk2 + lh][mw + lm];
      v2f a1 = As[cur][kk2 + lh][mw + 16 + lm];
      v2f b0 = Bs[cur][kk2 + lh][nw + lm];
      v2f b1 = Bs[cur][kk2 + lh][nw + 16 + lm];
      acc00 = wmma_f32(a0, b0, acc00);
      acc01 = wmma_f32(a0, b1, acc01);
      acc10 = wmma_f32(a1, b0, acc10);
      acc11 = wmma_f32(a1, b1, acc11);
    }
    if (more) {
      const int nxt = cur ^ 1;
      As[nxt][aKp+0][aRow] = mk2(ar0.x, ar0.y);
      As[nxt][aKp+1][aRow] = mk2(ar0.z, ar0.w);
      As[nxt][aKp+2][aRow] = mk2(ar1.x, ar1.y);
      As[nxt][aKp+3][aRow] = mk2(ar1.z, ar1.w);
      Bs[nxt][bKp+0][bN]   = mk2(br.x, br.y);
      Bs[nxt][bKp+1][bN]   = mk2(br.z, br.w);
    }
    __syncthreads();
  }

  // epilogue: C/D layout -> row = 8*lh + i within a 16x16 tile, col = lm
#pragma unroll
  for (int i = 0; i < 8; ++i) {
    int n0 = nTile + nw + lm;
    int n1 = n0 + 16;
    int mg0 = mTile + mw + lh*8 + i;
    if (mg0 < MTOT) {
      float v0 = acc00[i] + bias[n0];
      E[(size_t)mg0*DOUT + n0] = v0 > 0.f ? v0 : 0.f;
      float v1 = acc01[i] + bias[n1];
      E[(size_t)mg0*DOUT + n1] = v1 > 0.f ? v1 : 0.f;
    }
    int mg1 = mg0 + 16;
    if (mg1 < MTOT) {
      float v0 = acc10[i] + bias[n0];
      E[(size_t)mg1*DOUT + n0] = v0 > 0.f ? v0 : 0.f;
      float v1 = acc11[i] + bias[n1];
      E[(size_t)mg1*DOUT + n1] = v1 > 0.f ? v1 : 0.f;
    }
  }
}

// ============================================================================
// Kernel 2: row squared norms of E
// ============================================================================
__global__ __launch_bounds__(128)
void k_norm(const float* __restrict__ E, float* __restrict__ nrm)
{
  __shared__ float red[128];
  int r = blockIdx.x;
  const float* row = E + (size_t)r * DOUT;
  float s = 0.f;
  for (int i = threadIdx.x; i < DOUT; i += 128) { float v = row[i]; s += v*v; }
  red[threadIdx.x] = s; __syncthreads();
  for (int st = 64; st > 0; st >>= 1) {
    if (threadIdx.x < st) red[threadIdx.x] += red[threadIdx.x + st];
    __syncthreads();
  }
  if (threadIdx.x == 0) nrm[r] = red[0];
}

// ============================================================================
// Kernel 3: Dall[m, n] = sqrt(max(|E_m|^2 + |Es_n|^2 - 2 E_m.Es_n, 1e-12))
//   Es = E rows [MQ, MTOT). Same 128x64 double-buffered K-paired WMMA tiling.
// ============================================================================
__global__ __launch_bounds__(256)
void k_gram(const float* __restrict__ E, const float* __restrict__ nrm,
            float* __restrict__ Dall)
{
  __shared__ v2f As[2][8][130];
  __shared__ v2f Bs[2][8][66];
  const int tid   = threadIdx.x;
  const int mTile = blockIdx.x * 128;
  const int nTile = blockIdx.y * 64;

  const int aRow = tid >> 1;
  const int aKp  = (tid & 1) * 4;
  int mc = mTile + aRow; if (mc >= MTOT) mc = MTOT - 1;
  const float* arow = E + (size_t)mc * DOUT;

  const int bN  = tid & 63;
  const int bKp = (tid >> 6) * 2;
  int ncc = nTile + bN; if (ncc >= MS) ncc = MS - 1;
  const float* brow = E + (size_t)(MQ + ncc) * DOUT;

  const int wv   = tid >> 5;
  const int lane = tid & 31;
  const int lm   = lane & 15, lh = lane >> 4;
  const int mw   = (wv & 3) * 32;
  const int nw   = (wv >> 2) * 32;

  v8f acc00 = {}, acc01 = {}, acc10 = {}, acc11 = {};

  const int NSTAGE = DOUT / 16;       // 72

  {
    float4 ar0 = *(const float4*)(arow + aKp*2);
    float4 ar1 = *(const float4*)(arow + aKp*2 + 4);
    float4 br  = *(const float4*)(brow + bKp*2);
    As[0][aKp+0][aRow] = mk2(ar0.x, ar0.y);
    As[0][aKp+1][aRow] = mk2(ar0.z, ar0.w);
    As[0][aKp+2][aRow] = mk2(ar1.x, ar1.y);
    As[0][aKp+3][aRow] = mk2(ar1.z, ar1.w);
    Bs[0][bKp+0][bN]   = mk2(br.x, br.y);
    Bs[0][bKp+1][bN]   = mk2(br.z, br.w);
  }
  __syncthreads();

  for (int s = 0; s < NSTAGE; ++s) {
    const int cur = s & 1;
    float4 ar0, ar1, br;
    const bool more = (s + 1) < NSTAGE;
    if (more) {
      int k1 = (s + 1) * 16;
      ar0 = *(const float4*)(arow + k1 + aKp*2);
      ar1 = *(const float4*)(arow + k1 + aKp*2 + 4);
      br  = *(const float4*)(brow + k1 + bKp*2);
    }
#pragma unroll
    for (int kk2 = 0; kk2 < 8; kk2 += 2) {
      v2f a0 = As[cur][kk2 + lh][mw + lm];
      v2f a1 = As[cur][kk2 + lh][mw + 16 + lm];
      v2f b0 = Bs[cur][kk2 + lh][nw + lm];
      v2f b1 = Bs[cur][kk2 + lh][nw + 16 + lm];
      acc00 = wmma_f32(a0, b0, acc00);
      acc01 = wmma_f32(a0, b1, acc01);
      acc10 = wmma_f32(a1, b0, acc10);
      acc11 = wmma_f32(a1, b1, acc11);
    }
    if (more) {
      const int nxt = cur ^ 1;
      As[nxt][aKp+0][aRow] = mk2(ar0.x, ar0.y);
      As[nxt][aKp+1][aRow] = mk2(ar0.z, ar0.w);
      As[nxt][aKp+2][aRow] = mk2(ar1.x, ar1.y);
      As[nxt][aKp+3][aRow] = mk2(ar1.z, ar1.w);
      Bs[nxt][bKp+0][bN]   = mk2(br.x, br.y);
      Bs[nxt][bKp+1][bN]   = mk2(br.z, br.w);
    }
    __syncthreads();
  }

#pragma unroll
  for (int i = 0; i < 8; ++i) {
    int n0 = nTile + nw + lm;
    int n1 = n0 + 16;
    int mg0 = mTile + mw + lh*8 + i;
    int mg1 = mg0 + 16;
    if (mg0 < MTOT) {
      float nm = nrm[mg0];
      if (n0 < MS) {
        float d2 = nm + nrm[MQ + n0] - 2.f * acc00[i];
        Dall[(size_t)mg0*MS + n0] = sqrtf(fmaxf(d2, 1e-12f));
      }
      if (n1 < MS) {
        float d2 = nm + nrm[MQ + n1] - 2.f * acc01[i];
        Dall[(size_t)mg0*MS + n1] = sqrtf(fmaxf(d2, 1e-12f));
      }
    }
    if (mg1 < MTOT) {
      float nm = nrm[mg1];
      if (n0 < MS) {
        float d2 = nm + nrm[MQ + n0] - 2.f * acc10[i];
        Dall[(size_t)mg1*MS + n0] = sqrtf(fmaxf(d2, 1e-12f));
      }
      if (n1 < MS) {
        float d2 = nm + nrm[MQ + n1] - 2.f * acc11[i];
        Dall[(size_t)mg1*MS + n1] = sqrtf(fmaxf(d2, 1e-12f));
      }
    }
  }
}

// ============================================================================
// Kernel 4: per (query-tuple row r, class c):
//   rowmax = max_s D[r,s] ; pos = first argmax ;
//   ave    = mean_{j<16} max_{i<45} D[r, i*16+j]   (reference's reshape trick)
// ============================================================================
__global__ __launch_bounds__(128)
void k_rowred(const float* __restrict__ Dall, float* __restrict__ rowmax,
              int* __restrict__ pos, float* __restrict__ ave)
{
  __shared__ float sv[128];
  __shared__ int   si[128];
  __shared__ float gm[16];
  const int r = blockIdx.x, c = blockIdx.y;
  const float* row = Dall + (size_t)r*MS + (size_t)c*SSUP;

  float bv = -1.f; int bi = 0;
  for (int s = threadIdx.x; s < SSUP; s += 128) {
    float v = row[s];
    if (v > bv) { bv = v; bi = s; }
  }
  sv[threadIdx.x] = bv; si[threadIdx.x] = bi; __syncthreads();
  for (int st = 64; st > 0; st >>= 1) {
    if (threadIdx.x < st) {
      float ov = sv[threadIdx.x + st]; int oi = si[threadIdx.x + st];
      if (ov > sv[threadIdx.x] || (ov == sv[threadIdx.x] && oi < si[threadIdx.x])) {
        sv[threadIdx.x] = ov; si[threadIdx.x] = oi;
      }
    }
    __syncthreads();
  }
  if (threadIdx.x == 0) {
    rowmax[(size_t)c*MQ + r] = sv[0];
    pos   [(size_t)c*MQ + r] = si[0];
  }
  if (threadIdx.x < 16) {
    float g = -1.f;
    for (int i = 0; i < TT; ++i) { float v = row[i*16 + threadIdx.x]; g = v > g ? v : g; }
    gm[threadIdx.x] = g;
  }
  __syncthreads();
  if (threadIdx.x == 0) {
    float s = 0.f;
    for (int j = 0; j < 16; ++j) s += gm[j];
    ave[(size_t)c*MQ + r] = s * (1.f/16.f);
  }
}

// ============================================================================
// Kernel 5: rec[c,oi,j] = sum_r ( SS_dist[c][pos_c[r], o, j] > ave_c[r] )
// ============================================================================
__global__ __launch_bounds__(64)
void k_rec(const float* __restrict__ Dall, const int* __restrict__ pos,
           const float* __restrict__ ave, float* __restrict__ rec)
{
  __shared__ int   sp[64];
  __shared__ float sa[64];
  const int j  = blockIdx.x*64 + threadIdx.x;
  const int oi = blockIdx.y, c = blockIdx.z;
  const int o  = (oi < c) ? oi : oi + 1;        // OTHERS[c][oi]
  const float* base = Dall + (size_t)(MQ + c*SSUP)*MS + (size_t)o*SSUP;
  const bool valid = (j < SSUP);
  int cnt = 0;
  for (int r0 = 0; r0 < MQ; r0 += 64) {         // MQ % 64 == 0
    __syncthreads();
    sp[threadIdx.x] = pos[(size_t)c*MQ + r0 + threadIdx.x];
    sa[threadIdx.x] = ave[(size_t)c*MQ + r0 + threadIdx.x];
    __syncthreads();
    if (valid) {
      for (int i = 0; i < 64; ++i)
        cnt += (base[(size_t)sp[i]*MS + j] > sa[i]) ? 1 : 0;
    }
  }
  if (valid) rec[((size_t)c*(WAY-1) + oi)*SSUP + j] = (float)cnt;
}

// ============================================================================
// Kernel 6: per (c,oi): thr = sum(rec)/max(nnz,1) ; mask = rec < thr
// ============================================================================
__global__ __launch_bounds__(128)
void k_mask(const float* __restrict__ rec, float* __restrict__ mask)
{
  __shared__ float ssum[128];
  __shared__ float snz[128];
  const int oi = blockIdx.x, c = blockIdx.y;
  const float* rrow = rec + ((size_t)c*(WAY-1) + oi)*SSUP;
  float s = 0.f, nz = 0.f;
  for (int j = threadIdx.x; j < SSUP; j += 128) {
    float v = rrow[j]; s += v; nz += (v > 0.f) ? 1.f : 0.f;
  }
  ssum[threadIdx.x] = s; snz[threadIdx.x] = nz; __syncthreads();
  for (int st = 64; st > 0; st >>= 1) {
    if (threadIdx.x < st) {
      ssum[threadIdx.x] += ssum[threadIdx.x + st];
      snz[threadIdx.x]  += snz[threadIdx.x + st];
    }
    __syncthreads();
  }
  float thr = ssum[0] / fmaxf(snz[0], 1.f);
  for (int j = threadIdx.x; j < SSUP; j += 128)
    mask[((size_t)c*(WAY-1) + oi)*SSUP + j] = (rrow[j] < thr) ? 1.f : 0.f;
}

// Kernel 7: masksum[c] = sum over (WAY-1)*SSUP mask entries (scalar divisor)
__global__ __launch_bounds__(128)
void k_masksum(const float* __restrict__ mask, float* __restrict__ msum)
{
  __shared__ float red[128];
  const int c = blockIdx.x;
  float s = 0.f;
  for (int j = threadIdx.x; j < (WAY-1)*SSUP; j += 128)
    s += mask[(size_t)c*(WAY-1)*SSUP + j];
  red[threadIdx.x] = s; __syncthreads();
  for (int st = 64; st > 0; st >>= 1) {
    if (threadIdx.x < st) red[threadIdx.x] += red[threadIdx.x + st];
    __syncthreads();
  }
  if (threadIdx.x == 0) msum[c] = red[0];
}

// ============================================================================
// Kernel 8: crow[c,r] = sum_{oi,j} mask[c,oi,j] * Dall[r, OTHERS[c][oi]*720+j]
// ============================================================================
__global__ __launch_bounds__(128)
void k_contrast(const float* __restrict__ Dall, const float* __restrict__ mask,
                float* __restrict__ crow)
{
  __shared__ float red[128];
  const int r = blockIdx.x, c = blockIdx.y;
  const float* row = Dall + (size_t)r*MS;
  const float* mk  = mask + (size_t)c*(WAY-1)*SSUP;
  float s = 0.f;
  for (int jj = threadIdx.x; jj < (WAY-1)*SSUP; jj += 128) {
    int oi = jj / SSUP, j = jj - oi*SSUP;
    int o  = (oi < c) ? oi : oi + 1;
    s += mk[jj] * row[o*SSUP + j];
  }
  red[threadIdx.x] = s; __syncthreads();
  for (int st = 64; st > 0; st >>= 1) {
    if (threadIdx.x < st) red[threadIdx.x] += red[threadIdx.x + st];
    __syncthreads();
  }
  if (threadIdx.x == 0) crow[(size_t)c*MQ + r] = red[0];
}

// ============================================================================
// Kernel 9: final outputs  out[0] = dist_max [320,5] ; out[1] = logits_contrast
// ============================================================================
__global__ void k_final(const float* __restrict__ rowmax, const float* __restrict__ crow,
                        const float* __restrict__ msum, float* __restrict__ out)
{
  int idx = blockIdx.x*blockDim.x + threadIdx.x;
  if (idx >= NQ*WAY) return;
  int q = idx / WAY, c = idx % WAY;
  float dm = 0.f, ct = 0.f;
  for (int t = 0; t < TT; ++t) {
    dm += rowmax[(size_t)c*MQ + q*TT + t];
    ct += crow  [(size_t)c*MQ + q*TT + t];
  }
  dm *= (1.f/TT);
  ct  = ct / fmaxf(msum[c], 1.f) * (1.f/TT) * (1.f/(WAY-1));
  out[idx]         = dm;             // dist_max
  out[NQ*WAY+idx]  = dm / (dm + ct); // logits_contrast
}

// ============================================================================
extern "C" void kernel_launch(void* const* d_in, const int* in_sizes, int n_in,
                              void* d_out, int out_size, void* d_ws, size_t ws_size,
                              hipStream_t stream)
{
  const float* sup  = (const float*)d_in[0];  // [80,10,2048]
  const float* qry  = (const float*)d_in[1];  // [320,10,2048]
  // d_in[2] support_labels: repeat(arange(5),16) is sorted -> argsort is identity
  const float* W    = (const float*)d_in[3];  // [1152,4096]
  const float* bias = (const float*)d_in[4];  // [1152]

  // workspace carve-up (~344 MB of f32, assumed <= ws_size)
  float* ws = (float*)d_ws;
  size_t off = 0;
  float* E      = ws + off; off += (size_t)MTOT*DOUT;   // embeddings
  float* nrm    = ws + off; off += MTOT;                // row |.|^2
  float* Dall   = ws + off; off += (size_t)MTOT*MS;     // all distances
  float* ave    = ws + off; off += (size_t)WAY*MQ;
  float* rowmax = ws + off; off += (size_t)WAY*MQ;
  int*   pos    = (int*)(ws + off); off += (size_t)WAY*MQ;
  float* rec    = ws + off; off += (size_t)WAY*(WAY-1)*SSUP;
  float* mask   = ws + off; off += (size_t)WAY*(WAY-1)*SSUP;
  float* msum   = ws + off; off += 8;
  float* crow   = ws + off; off += (size_t)WAY*MQ;
  (void)in_sizes; (void)n_in; (void)out_size; (void)ws_size;

  dim3 b256(256);
  hipLaunchKernelGGL(k_embed, dim3((MTOT+127)/128, DOUT/64), b256, 0, stream,
                     sup, qry, W, bias, E);
  hipLaunchKernelGGL(k_norm, dim3(MTOT), dim3(128), 0, stream, E, nrm);
  hipLaunchKernelGGL(k_gram, dim3((MTOT+127)/128, (MS+63)/64), b256, 0, stream,
                     E, nrm, Dall);
  hipLaunchKernelGGL(k_rowred, dim3(MQ, WAY), dim3(128), 0, stream,
                     Dall, rowmax, pos, ave);
  hipLaunchKernelGGL(k_rec, dim3((SSUP+63)/64, WAY-1, WAY), dim3(64), 0, stream,
                     Dall, pos, ave, rec);
  hipLaunchKernelGGL(k_mask, dim3(WAY-1, WAY), dim3(128), 0, stream, rec, mask);
  hipLaunchKernelGGL(k_masksum, dim3(WAY), dim3(128), 0, stream, mask, msum);
  hipLaunchKernelGGL(k_contrast, dim3(MQ, WAY), dim3(128), 0, stream,
                     Dall, mask, crow);
  hipLaunchKernelGGL(k_final, dim3((NQ*WAY+255)/256), dim3(256), 0, stream,
                     rowmax, crow, msum, (float*)d_out);
}